// GatedTSNorm_70257075028071
// MI455X (gfx1250) — compile-verified
//
#include <hip/hip_runtime.h>
#include <hip/hip_bf16.h>
#include <math.h>

#define MOMENTUM 0.05f
#define EPS_V 1e-6f

typedef float v2f __attribute__((ext_vector_type(2)));
typedef float v4f __attribute__((ext_vector_type(4)));
typedef float v8f __attribute__((ext_vector_type(8)));

// ---------------------------------------------------------------------------
// Kernel 0: softmax over C for Wa_w (block 0) and Wb_w (block 1).
// ---------------------------------------------------------------------------
__global__ void __launch_bounds__(512)
softmax_weights(const float* __restrict__ Wa, const float* __restrict__ Wb,
                float* __restrict__ wa, float* __restrict__ wb, int C) {
    __shared__ float red[1024];
    int tid = threadIdx.x;
    const float* src = (blockIdx.x == 0) ? Wa : Wb;
    float*       dst = (blockIdx.x == 0) ? wa : wb;
    float v = (tid < C) ? src[tid] : -INFINITY;
    red[tid] = v; __syncthreads();
    for (int off = blockDim.x >> 1; off > 0; off >>= 1) {
        if (tid < off) red[tid] = fmaxf(red[tid], red[tid + off]);
        __syncthreads();
    }
    float mx = red[0]; __syncthreads();
    float e = (tid < C) ? __expf(v - mx) : 0.f;
    red[tid] = e; __syncthreads();
    for (int off = blockDim.x >> 1; off > 0; off >>= 1) {
        if (tid < off) red[tid] += red[tid + off];
        __syncthreads();
    }
    float s = red[0];
    if (tid < C) dst[tid] = e / s;
}

// ---------------------------------------------------------------------------
// Kernel 1: channel reduction via V_WMMA_F32_16X16X4_F32.
// One wave per 16-timestep tile.  D = A(weights 16x4) * B(x 4x16), accumulated
// over C/4 chunks.  acc1 row0 = xa (wa.x), row1 = xb (wb.x); acc2 row0 = xq
// (wb.x^2).
// A-operand fetch is select-free: each lane owns an LDS pointer (wa / wb /
// zero-slot) with a per-lane stride (4 floats or 0), so the inner loop is just
// 2x ds_load_b64 + 2x global_load_b32 + 2x v_mul + 2x v_wmma, EXEC all-ones.
// ---------------------------------------------------------------------------
__global__ void __launch_bounds__(256)
reduce_wmma(const float* __restrict__ x,
            const float* __restrict__ wa, const float* __restrict__ wb,
            float* __restrict__ xa, float* __restrict__ xb,
            float* __restrict__ xq, int C, int T) {
    __shared__ float ldsw[1024 + 8];                // [0,C)=wa, [C,2C)=wb, [2C,2C+8)=0
    int tid = threadIdx.x;
    for (int i = tid; i < 2 * C; i += blockDim.x)
        ldsw[i] = (i < C) ? wa[i] : wb[i - C];
    if (tid < 8) ldsw[2 * C + tid] = 0.f;
    __syncthreads();

    int lane = tid & 31;
    int wave = blockIdx.x * (blockDim.x >> 5) + (tid >> 5);
    int tilesPerB = T >> 4;
    int b  = wave / tilesPerB;
    int t0 = (wave % tilesPerB) << 4;

    int  m    = lane & 15;                          // A: row (weight id)  B: col (t)
    int  koff = (lane >> 4) << 1;                   // K pair held by this half-wave
    bool m0 = (m == 0), m1 = (m == 1);

    // Per-lane LDS pointers + strides (set up once, no loop divergence).
    const float* a1p = m0 ? &ldsw[koff] : (m1 ? &ldsw[C + koff] : &ldsw[2 * C]);
    const float* a2p = m0 ? &ldsw[C + koff] : &ldsw[2 * C];
    int a1s = (m < 2) ? 4 : 0;
    int a2s = m0 ? 4 : 0;

    v8f acc1 = {}; v8f acc2 = {};
    const float* xp = x + (size_t)b * C * T + (size_t)koff * T + t0 + m;
    const size_t xstep = 4 * (size_t)T;

    for (int c0 = 0; c0 < C; c0 += 4) {
        v2f A1 = *(const v2f*)a1p;                  // ds_load_b64
        v2f A2 = *(const v2f*)a2p;                  // ds_load_b64
        v2f Bx;
        Bx.x = xp[0];
        Bx.y = xp[(size_t)T];
        v2f Bxx = Bx * Bx;
        acc1 = __builtin_amdgcn_wmma_f32_16x16x4_f32(false, A1, false, Bx,
                                                     (short)0, acc1, false, false);
        acc2 = __builtin_amdgcn_wmma_f32_16x16x4_f32(false, A2, false, Bxx,
                                                     (short)0, acc2, false, false);
        a1p += a1s;
        a2p += a2s;
        xp  += xstep;
    }
    if (lane < 16) {                                // rows 0/1 -> v[0],v[1], lanes 0-15
        size_t o = (size_t)b * T + t0 + lane;
        xa[o] = acc1[0];
        xb[o] = acc1[1];
        xq[o] = acc2[0];
    }
}

// ---------------------------------------------------------------------------
// Kernel 2: two gated-EMA scans per batch (mean, then var -> invstd).
// y[t] = a[t]*y[t-1] + b[t];  composition (a2,b2)o(a1,b1) = (a2*a1, a2*b1+b2).
// 512 threads * 16 elems = T.  Serial compose + Hillis-Steele block scan.
// ---------------------------------------------------------------------------
#define SCAN_THREADS 512
#define SCAN_E 16
__global__ void __launch_bounds__(SCAN_THREADS)
gated_scan(const float* __restrict__ g,
           const float* __restrict__ xa, const float* __restrict__ xb,
           const float* __restrict__ xq,
           float* __restrict__ mean, float* __restrict__ invstd, int T) {
    __shared__ float sA[SCAN_THREADS], sB[SCAN_THREADS];
    int b = blockIdx.x, tid = threadIdx.x;
    size_t base = (size_t)b * T + (size_t)tid * SCAN_E;

    float a_loc[SCAN_E], b_loc[SCAN_E], m_loc[SCAN_E];

    // ---- pass 1: mean = EMA(xa) ----
    float A = 1.f, Bc = 0.f;
    for (int i = 0; i < SCAN_E; ++i) {
        float gt = g[base + i] * MOMENTUM;
        float a  = 1.f - gt;
        float bb = gt * xa[base + i];
        a_loc[i] = a; b_loc[i] = bb;
        Bc = a * Bc + bb;  A *= a;
    }
    sA[tid] = A; sB[tid] = Bc; __syncthreads();
    for (int off = 1; off < SCAN_THREADS; off <<= 1) {
        float pA = 1.f, pB = 0.f;
        if (tid >= off) { pA = sA[tid - off]; pB = sB[tid - off]; }
        __syncthreads();
        if (tid >= off) { float Ac = sA[tid]; sB[tid] = Ac * pB + sB[tid]; sA[tid] = Ac * pA; }
        __syncthreads();
    }
    float y = (tid == 0) ? 0.f : sB[tid - 1];
    for (int i = 0; i < SCAN_E; ++i) {
        y = a_loc[i] * y + b_loc[i];
        m_loc[i] = y;
        mean[base + i] = y;
    }
    __syncthreads();

    // ---- pass 2: var = EMA(xq - 2*mean*xb + mean^2), same gates ----
    A = 1.f; Bc = 0.f;
    for (int i = 0; i < SCAN_E; ++i) {
        float mv = m_loc[i];
        float s  = xq[base + i] - 2.f * mv * xb[base + i] + mv * mv;
        float gt = 1.f - a_loc[i];
        float bb = gt * s;
        b_loc[i] = bb;
        Bc = a_loc[i] * Bc + bb;  A *= a_loc[i];
    }
    sA[tid] = A; sB[tid] = Bc; __syncthreads();
    for (int off = 1; off < SCAN_THREADS; off <<= 1) {
        float pA = 1.f, pB = 0.f;
        if (tid >= off) { pA = sA[tid - off]; pB = sB[tid - off]; }
        __syncthreads();
        if (tid >= off) { float Ac = sA[tid]; sB[tid] = Ac * pB + sB[tid]; sA[tid] = Ac * pA; }
        __syncthreads();
    }
    y = (tid == 0) ? 0.f : sB[tid - 1];
    for (int i = 0; i < SCAN_E; ++i) {
        y = a_loc[i] * y + b_loc[i];
        invstd[base + i] = rsqrtf(y + EPS_V);
    }
}

// ---------------------------------------------------------------------------
// Kernel 3: out = (x - mean[b,t]) * invstd[b,t] * Wo_w[c] + Wo_b[c]
// 2-D grid: grid.y = b*C rows (row decomposition is scalar), grid.x tiles T/4.
// x read NT (last use; L2-resident from kernel 1); out stored NT.
// ---------------------------------------------------------------------------
__global__ void __launch_bounds__(256)
apply_norm(const float* __restrict__ x,
           const float* __restrict__ mean, const float* __restrict__ invstd,
           const float* __restrict__ Wo_w, const float* __restrict__ Wo_b,
           float* __restrict__ out, int C, int T) {
    int row = blockIdx.y;                               // b*C + c (uniform)
    int c   = row % C;
    int b   = row / C;
    int t4  = (blockIdx.x * blockDim.x + threadIdx.x) * 4;

    size_t xo = (size_t)row * T + t4;
    size_t mo = (size_t)b   * T + t4;

    v4f xv = __builtin_nontemporal_load((const v4f*)(x + xo));
    v4f mv = *(const v4f*)(mean   + mo);
    v4f iv = *(const v4f*)(invstd + mo);
    float w  = Wo_w[c];
    float bi = Wo_b[c];
    v4f o;
    o.x = (xv.x - mv.x) * iv.x * w + bi;
    o.y = (xv.y - mv.y) * iv.y * w + bi;
    o.z = (xv.z - mv.z) * iv.z * w + bi;
    o.w = (xv.w - mv.w) * iv.w * w + bi;
    __builtin_nontemporal_store(o, (v4f*)(out + xo));
}

// ---------------------------------------------------------------------------
extern "C" void kernel_launch(void* const* d_in, const int* in_sizes, int n_in,
                              void* d_out, int out_size, void* d_ws, size_t ws_size,
                              hipStream_t stream) {
    const float* x    = (const float*)d_in[0];
    const float* g    = (const float*)d_in[1];
    const float* Wa_w = (const float*)d_in[2];
    const float* Wb_w = (const float*)d_in[3];
    const float* Wo_w = (const float*)d_in[4];
    const float* Wo_b = (const float*)d_in[5];
    float* out = (float*)d_out;

    const int C  = in_sizes[2];         // 512
    const int BT = in_sizes[1];         // B*T
    const int T  = 8192;                // reference shape
    const int B  = BT / T;              // 8

    float* ws   = (float*)d_ws;
    float* wa   = ws;                   // C
    float* wb   = wa + C;               // C
    float* xa   = wb + C;               // B*T
    float* xb   = xa + BT;              // B*T
    float* xq   = xb + BT;              // B*T
    float* mean = xq + BT;              // B*T
    float* inv  = mean + BT;            // B*T

    softmax_weights<<<2, 512, 0, stream>>>(Wa_w, Wb_w, wa, wb, C);

    int waves  = (B * T) / 16;          // one wave per 16-t tile
    int blocks = waves / 8;             // 8 waves (256 thr) per block
    reduce_wmma<<<blocks, 256, 0, stream>>>(x, wa, wb, xa, xb, xq, C, T);

    gated_scan<<<B, SCAN_THREADS, 0, stream>>>(g, xa, xb, xq, mean, inv, T);

    dim3 agrid(T / (256 * 4), B * C);
    apply_norm<<<agrid, 256, 0, stream>>>(x, mean, inv, Wo_w, Wo_b, out, C, T);
}